// AttentiveReduce_18133351923879
// MI455X (gfx1250) — compile-verified
//
#include <hip/hip_runtime.h>
#include <math.h>

#define NEG_SLOPE 0.2f
#define DIM   128
#define LDSS  132      // padded LDS row stride (floats): lane m -> bank 4m, conflict-free float2
#define CHUNK 64

typedef float v2f __attribute__((ext_vector_type(2)));
typedef float v8f __attribute__((ext_vector_type(8)));

// ---------------------------------------------------------------------------
// Kernel 1: exclusive prefix sum of segment sizes -> segment start offsets.
// Single workgroup, 1024 threads, 16 elements/thread (covers B = 16384).
// ---------------------------------------------------------------------------
__global__ __launch_bounds__(1024)
void seg_offsets_kernel(const int* __restrict__ sizes, int* __restrict__ offs, int B) {
    __shared__ int part[1024];
    const int t = threadIdx.x;
    int local[16];
    int sum = 0;
    const int base = t * 16;
#pragma unroll
    for (int j = 0; j < 16; ++j) {
        const int idx = base + j;
        const int v = (idx < B) ? sizes[idx] : 0;
        local[j] = v;
        sum += v;
    }
    part[t] = sum;
    __syncthreads();
    for (int off = 1; off < 1024; off <<= 1) {
        const int v   = part[t];
        const int add = (t >= off) ? part[t - off] : 0;
        __syncthreads();
        part[t] = v + add;
        __syncthreads();
    }
    int run = (t > 0) ? part[t - 1] : 0;
#pragma unroll
    for (int j = 0; j < 16; ++j) {
        const int idx = base + j;
        if (idx < B) offs[idx] = run;
        run += local[j];
    }
}

// ---------------------------------------------------------------------------
// Kernel 2: one workgroup per segment (128 threads = 4 wave32s).
//   - stage segment tile into LDS (register-buffered, 16 loads in flight)
//   - scores via V_WMMA_F32_16X16X4_F32 (wave w: rows [16w,16w+16))
//   - softmax reductions on wave 0 only (shfl_xor), broadcast 2 scalars
//   - per-thread feature accumulator readout, online-softmax rescaled
// ---------------------------------------------------------------------------
__global__ __launch_bounds__(128)
void attentive_reduce_kernel(const float* __restrict__ feat,
                             const int*   __restrict__ sizes,
                             const float* __restrict__ w,
                             const int*   __restrict__ offs,
                             float*       __restrict__ out) {
    __shared__ float tile[CHUNK * LDSS];   // 33,792 B segment tile (padded rows)
    __shared__ float wlds[DIM];
    __shared__ float sc[CHUNK];
    __shared__ float el[CHUNK];
    __shared__ float bc[2];                // {newM, sumE} broadcast

    const int b    = blockIdx.x;
    const int t    = threadIdx.x;
    const int lane = t & 31;
    const int wv   = t >> 5;      // wave id 0..3
    const int m    = lane & 15;   // row within 16-row WMMA tile
    const int h    = lane >> 4;   // K half-group (A layout: lanes 16-31 hold K=2,3)

    const int start = offs[b];
    const int n     = sizes[b];

    wlds[t] = w[t];               // 128 threads load the 128-long weight vector

    float M = -INFINITY, S = 0.f, acc = 0.f;

    const int row0 = t >> 5;          // staging: fixed row base per thread
    const int c4   = (t & 31) << 2;   // staging: fixed column (float4 granule)

    for (int c0 = 0; c0 < n; c0 += CHUNK) {
        const int nc = min(CHUNK, n - c0);

        // ---- stage chunk rows into LDS ----
        if (nc == CHUNK) {
            // fast path: 16 loads issued back-to-back (full MLP), then store
            float4 buf[16];
            const float* gbase = feat + ((size_t)(start + c0 + row0)) * DIM + c4;
#pragma unroll
            for (int j = 0; j < 16; ++j)
                buf[j] = *(const float4*)(gbase + (size_t)j * 4 * DIM);
            float* lbase = &tile[row0 * LDSS + c4];
#pragma unroll
            for (int j = 0; j < 16; ++j)
                *(float4*)(lbase + j * 4 * LDSS) = buf[j];
        } else {
            for (int v = t; v < nc * (DIM / 4); v += 128) {
                const int row = v >> 5;
                const int cc  = (v & 31) << 2;
                const float4 f4 =
                    *(const float4*)(feat + ((size_t)(start + c0 + row)) * DIM + cc);
                *(float4*)(&tile[row * LDSS + cc]) = f4;
            }
        }
        __syncthreads();

        // ---- scores: D(16x16) = A(16x4,f32) x B(4x16,f32) over K=128 ----
        // B[k][n] = w[k] broadcast across N columns; only column 0 is used.
        {
            const int rb = wv * 16;
            v8f c = {};
            const float* arow = &tile[(rb + m) * LDSS];
#pragma unroll
            for (int k0 = 0; k0 < DIM; k0 += 4) {
                v2f a, bm;
                a.x  = arow[k0 + 2 * h];
                a.y  = arow[k0 + 2 * h + 1];
                bm.x = wlds[k0 + 2 * h];
                bm.y = wlds[k0 + 2 * h + 1];
                c = __builtin_amdgcn_wmma_f32_16x16x4_f32(
                        false, a, false, bm, (short)0, c, false, false);
            }
            // C/D layout: VGPR r holds M=r (lanes 0-15) / M=r+8 (lanes 16-31).
            // Column N=0 lives in lanes 0 and 16.
            if (m == 0) {
#pragma unroll
                for (int r = 0; r < 8; ++r) sc[rb + h * 8 + r] = c[r];
            }
        }
        __syncthreads();

        // ---- softmax bookkeeping: wave 0 only, shuffle reductions ----
        if (wv == 0) {
            float s0 = (lane      < nc) ? sc[lane]      : -INFINITY;
            float s1 = (lane + 32 < nc) ? sc[lane + 32] : -INFINITY;
            s0 = (s0 >= 0.f) ? s0 : NEG_SLOPE * s0;
            s1 = (s1 >= 0.f) ? s1 : NEG_SLOPE * s1;
            float mx = fmaxf(s0, s1);
#pragma unroll
            for (int off = 16; off > 0; off >>= 1)
                mx = fmaxf(mx, __shfl_xor(mx, off, 32));
            const float newM = fmaxf(M, mx);
            const float e0 = (lane      < nc) ? __expf(s0 - newM) : 0.f;
            const float e1 = (lane + 32 < nc) ? __expf(s1 - newM) : 0.f;
            el[lane]      = e0;
            el[lane + 32] = e1;
            float sm = e0 + e1;
#pragma unroll
            for (int off = 16; off > 0; off >>= 1)
                sm += __shfl_xor(sm, off, 32);
            if (lane == 0) { bc[0] = newM; bc[1] = sm; }
        }
        __syncthreads();

        // ---- per-thread online-softmax update + weighted readout ----
        const float newM  = bc[0];
        const float sumE  = bc[1];
        const float scale = __expf(M - newM);
        S   = S * scale + sumE;
        acc = acc * scale;
        if (nc == CHUNK) {
#pragma unroll 16
            for (int i = 0; i < CHUNK; ++i)
                acc = fmaf(el[i], tile[i * LDSS + t], acc);
        } else {
            for (int i = 0; i < nc; ++i)
                acc = fmaf(el[i], tile[i * LDSS + t], acc);
        }
        M = newM;
        __syncthreads();   // tile/sc/el reused next chunk
    }

    out[(size_t)b * DIM + t] = (S > 0.f) ? (acc / S) : 0.f;
}

// ---------------------------------------------------------------------------
extern "C" void kernel_launch(void* const* d_in, const int* in_sizes, int n_in,
                              void* d_out, int out_size, void* d_ws, size_t ws_size,
                              hipStream_t stream) {
    const float* feat  = (const float*)d_in[0];
    const int*   sizes = (const int*)d_in[1];
    const float* w     = (const float*)d_in[2];
    float*       out   = (float*)d_out;
    const int B = in_sizes[1];          // number of segments (16384)
    int* offs = (int*)d_ws;             // B ints of scratch

    seg_offsets_kernel<<<1, 1024, 0, stream>>>(sizes, offs, B);
    attentive_reduce_kernel<<<B, 128, 0, stream>>>(feat, sizes, w, offs, out);
}